// AttentionSampler_68161130987974
// MI455X (gfx1250) — compile-verified
//
#include <hip/hip_runtime.h>
#include <math.h>

// softmax(sites @ w_site + c) == softmax(sites @ w_site): the scalar term
// (local . w_local + b) is shift-invariant under softmax, so we drop it.

typedef __attribute__((ext_vector_type(2))) float v2f;
typedef __attribute__((ext_vector_type(8))) float v8f;

#define DD 256
#define NTOT 500000
#define NTILES (NTOT / 16)   // 31250, exact
#define NB_RED 256

// ---------------------------------------------------------------------------
// K1: energies[i] = sites[i,:] . w_site  via V_WMMA_F32_16X16X4_F32.
// One wave handles a 16-row tile; 64 wmma steps walk K=0..255 in chunks of 4.
// A layout (16x4 f32): lanes 0-15 hold (K=k0,k0+1) of rows 0-15,
//                      lanes 16-31 hold (K=k0+2,k0+3) of the same rows.
// B layout (4x16 f32): VGPR v holds B-row K=v (lanes 0-15) / K=v+2 (16-31);
// we replicate w across all 16 columns so every column of D equals energies.
// ---------------------------------------------------------------------------
__global__ __launch_bounds__(256) void energies_wmma_kernel(
    const float* __restrict__ sites,
    const float* __restrict__ attn_w,   // w_site = attn_w + DD
    float* __restrict__ energies)
{
    __shared__ float w_lds[DD];
    const int tid = threadIdx.x;
    w_lds[tid] = attn_w[DD + tid];      // stage w_site in LDS (1 KB)
    __syncthreads();

    const int lane = tid & 31;
    const int hi   = lane >> 4;         // 0: lanes 0-15, 1: lanes 16-31
    const int lo   = lane & 15;
    const int wavesPerBlock = blockDim.x >> 5;
    const int waveId     = blockIdx.x * wavesPerBlock + (tid >> 5);
    const int waveStride = gridDim.x * wavesPerBlock;

    for (int t = waveId; t < NTILES; t += waveStride) {
        const int rowBase = t << 4;
        // this lane streams half of its row: bytes [4*(k0+2*hi) .. 4*(k0+2*hi)+8) per step
        const float* aPtr = sites + (size_t)(rowBase + lo) * DD + 2 * hi;
        v8f acc = {};
        #pragma unroll 8
        for (int k0 = 0; k0 < DD; k0 += 4) {
            v2f a = *(const v2f*)(aPtr + k0);                 // global_load_b64
            v2f b = *(const v2f*)(&w_lds[k0 + 2 * hi]);       // ds_load_b64
            acc = __builtin_amdgcn_wmma_f32_16x16x4_f32(
                /*neg_a=*/false, a, /*neg_b=*/false, b,
                /*c_mod=*/(short)0, acc,
                /*reuse_a=*/false, /*reuse_b=*/false);
        }
        // D: VGPR v holds row (v + 8*hi); all columns identical -> col 0 lanes
        if (lo == 0) {
            float* e = energies + rowBase + hi * 8;
            #pragma unroll
            for (int v = 0; v < 8; ++v) e[v] = acc[v];
        }
    }
}

// ---------------------------------------------------------------------------
// Reductions over the 2 MB energies vector (L2-resident; cost is negligible
// next to the 512 MB streaming pass).
// ---------------------------------------------------------------------------
__global__ __launch_bounds__(256) void block_max_kernel(
    const float* __restrict__ e, float* __restrict__ partial, int n)
{
    __shared__ float red[256];
    float m = -INFINITY;
    for (int i = blockIdx.x * blockDim.x + threadIdx.x; i < n;
         i += gridDim.x * blockDim.x)
        m = fmaxf(m, e[i]);
    red[threadIdx.x] = m;
    __syncthreads();
    for (int s = 128; s > 0; s >>= 1) {
        if (threadIdx.x < s)
            red[threadIdx.x] = fmaxf(red[threadIdx.x], red[threadIdx.x + s]);
        __syncthreads();
    }
    if (threadIdx.x == 0) partial[blockIdx.x] = red[0];
}

__global__ __launch_bounds__(256) void exp_sum_kernel(
    float* __restrict__ out, const float* __restrict__ gmax,
    float* __restrict__ partial, int n)
{
    __shared__ float red[256];
    const float M = *gmax;
    float s = 0.0f;
    for (int i = blockIdx.x * blockDim.x + threadIdx.x; i < n;
         i += gridDim.x * blockDim.x) {
        float v = __expf(out[i] - M);
        out[i] = v;
        s += v;
    }
    red[threadIdx.x] = s;
    __syncthreads();
    for (int st = 128; st > 0; st >>= 1) {
        if (threadIdx.x < st) red[threadIdx.x] += red[threadIdx.x + st];
        __syncthreads();
    }
    if (threadIdx.x == 0) partial[blockIdx.x] = red[0];
}

__global__ __launch_bounds__(256) void final_sum_inv_kernel(
    const float* __restrict__ partial, float* __restrict__ ginv, int n)
{
    __shared__ float red[256];
    float s = (threadIdx.x < n) ? partial[threadIdx.x] : 0.0f;
    red[threadIdx.x] = s;
    __syncthreads();
    for (int st = 128; st > 0; st >>= 1) {
        if (threadIdx.x < st) red[threadIdx.x] += red[threadIdx.x + st];
        __syncthreads();
    }
    if (threadIdx.x == 0) *ginv = 1.0f / red[0];
}

__global__ __launch_bounds__(256) void scale_kernel(
    float* __restrict__ out, const float* __restrict__ ginv, int n)
{
    const float inv = *ginv;
    for (int i = blockIdx.x * blockDim.x + threadIdx.x; i < n;
         i += gridDim.x * blockDim.x)
        out[i] *= inv;
}

extern "C" void kernel_launch(void* const* d_in, const int* in_sizes, int n_in,
                              void* d_out, int out_size, void* d_ws, size_t ws_size,
                              hipStream_t stream)
{
    // inputs: 0=local (unused), 1=sites, 2=attn_w, 3=attn_b (unused)
    const float* sites  = (const float*)d_in[1];
    const float* attn_w = (const float*)d_in[2];
    float* out = (float*)d_out;

    float* ws    = (float*)d_ws;
    float* gmax  = ws + 0;
    float* ginv  = ws + 1;
    float* pmax  = ws + 16;            // NB_RED floats
    float* psum  = ws + 16 + NB_RED;   // NB_RED floats

    // 1) matvec via WMMA, energies into d_out (512 MB streaming pass)
    energies_wmma_kernel<<<512, 256, 0, stream>>>(sites, attn_w, out);
    // 2) global max (exact, order-independent)
    block_max_kernel<<<NB_RED, 256, 0, stream>>>(out, pmax, NTOT);
    block_max_kernel<<<1, 256, 0, stream>>>(pmax, gmax, NB_RED);
    // 3) exp(e - max) in place + partial sums (fixed tree -> deterministic)
    exp_sum_kernel<<<NB_RED, 256, 0, stream>>>(out, gmax, psum, NTOT);
    final_sum_inv_kernel<<<1, 256, 0, stream>>>(psum, ginv, NB_RED);
    // 4) normalize
    scale_kernel<<<NB_RED, 256, 0, stream>>>(out, ginv, NTOT);
}